// AutoFM_21835613733414
// MI455X (gfx1250) — compile-verified
//
#include <hip/hip_runtime.h>
#include <hip/hip_bf16.h>

typedef __attribute__((ext_vector_type(2))) float v2f;
typedef __attribute__((ext_vector_type(8))) float v8f;

#define BSZ   4096
#define FDIM  39
#define EDIM  40
#define NPAIR 741          // 39*38/2
#define RPAD  48           // 3 tiles of 16 rows
#define WAVES_PER_BLOCK 8

// ---------------------------------------------------------------------------
// Kernel 1: per-sample Gram matrix via V_WMMA_F32_16X16X4_F32.
// One wave32 per sample; 8 samples per 256-thread block.
// ---------------------------------------------------------------------------
__global__ __launch_bounds__(256) void fm_gram_kernel(
    const int*   __restrict__ inputs,   // B*F indices
    const float* __restrict__ w,        // V
    const float* __restrict__ v,        // V*E
    float* __restrict__ level2,         // B*NPAIR (out)
    float* __restrict__ lsum)           // B (out)
{
    __shared__ float sX[WAVES_PER_BLOCK][RPAD * EDIM];

    const int wave = threadIdx.x >> 5;
    const int lane = threadIdx.x & 31;
    const int s    = blockIdx.x * WAVES_PER_BLOCK + wave;

    float* X = sX[wave];
    const int* myidx = inputs + s * FDIM;

    // Zero the padding rows 39..47 so WMMA on padded tiles contributes 0.
    for (int e = FDIM * EDIM + lane; e < RPAD * EDIM; e += 32)
        X[e] = 0.0f;

    // Gather the 39x40 embedding block for this sample into LDS.
    for (int e = lane; e < FDIM * EDIM; e += 32) {
        int row = e / EDIM;
        int col = e - row * EDIM;
        int idx = myidx[row];
        X[e] = v[(size_t)idx * EDIM + col];
    }

    // Linear term: l[s] = sum_f w[inputs[s,f]]
    if (lane == 0) {
        float acc = 0.0f;
        for (int f = 0; f < FDIM; ++f) acc += w[myidx[f]];
        lsum[s] = acc;
    }

    __syncthreads();

    const int half = lane >> 4;      // 0: K+{0,1}, 1: K+{2,3}
    const int mrow = lane & 15;      // M (for A) / N (for B) within tile
    float* out = level2 + (size_t)s * NPAIR;

    #pragma unroll
    for (int ti = 0; ti < 3; ++ti) {
        // Preload A operands for this tile row (reused across tj).
        v2f aK[10];
        const float* arow = X + (ti * 16 + mrow) * EDIM + half * 2;
        #pragma unroll
        for (int k = 0; k < 10; ++k)
            aK[k] = *(const v2f*)(arow + k * 4);

        #pragma unroll
        for (int tj = ti; tj < 3; ++tj) {
            const float* brow = X + (tj * 16 + mrow) * EDIM + half * 2;
            v8f c = {};
            #pragma unroll
            for (int k = 0; k < 10; ++k) {
                v2f b = *(const v2f*)(brow + k * 4);
                // 8 args: (neg_a, A, neg_b, B, c_mod, C, reuse_a, reuse_b)
                c = __builtin_amdgcn_wmma_f32_16x16x4_f32(
                        false, aK[k], false, b, (short)0, c, false, false);
            }
            // D layout: VGPR r, lane L -> M = r + 8*(L>=16), N = L&15.
            const int gj = tj * 16 + mrow;   // global column (larger index j)
            #pragma unroll
            for (int r = 0; r < 8; ++r) {
                int gi = ti * 16 + r + half * 8;   // global row (smaller index i)
                if (gi < gj && gj < FDIM) {
                    // combinations(F,2) order: p = i*(2F-i-1)/2 + (j-i-1)
                    int p = gi * (2 * FDIM - gi - 1) / 2 + (gj - gi - 1);
                    out[p] = c[r];
                }
            }
        }
    }
}

// ---------------------------------------------------------------------------
// Kernel 2: per-pair batch statistics -> fused scale/shift.
// Thread-per-pair; loads are coalesced across the pair dimension.
// ---------------------------------------------------------------------------
__global__ __launch_bounds__(256) void fm_bnstat_kernel(
    const float* __restrict__ level2,   // B*NPAIR
    const float* __restrict__ ew,       // NPAIR edge weights
    float* __restrict__ scale,          // NPAIR (out)
    float* __restrict__ shift)          // NPAIR (out)
{
    int p = blockIdx.x * 256 + threadIdx.x;
    if (p >= NPAIR) return;
    float s = 0.0f, s2 = 0.0f;
    for (int i = 0; i < BSZ; ++i) {
        float x = level2[(size_t)i * NPAIR + p];
        s += x; s2 += x * x;
    }
    const float invB = 1.0f / (float)BSZ;
    float mean = s * invB;
    float var  = s2 * invB - mean * mean;
    float sc   = rsqrtf(var + 1e-3f) * ew[p];
    scale[p] = sc;
    shift[p] = -mean * sc;
}

// ---------------------------------------------------------------------------
// Kernel 3: per-sample weighted reduction + linear term + bias.
// Block-per-sample; folds the shift sum into the same pass.
// ---------------------------------------------------------------------------
__global__ __launch_bounds__(256) void fm_out_kernel(
    const float* __restrict__ level2,
    const float* __restrict__ scale,
    const float* __restrict__ shift,
    const float* __restrict__ lsum,
    const float* __restrict__ bptr,
    float* __restrict__ out)
{
    __shared__ float red[256];
    const int s = blockIdx.x;
    const float* row = level2 + (size_t)s * NPAIR;
    float acc = 0.0f;
    for (int p = threadIdx.x; p < NPAIR; p += 256)
        acc += fmaf(row[p], scale[p], shift[p]);
    red[threadIdx.x] = acc;
    __syncthreads();
    for (int off = 128; off > 0; off >>= 1) {
        if (threadIdx.x < off) red[threadIdx.x] += red[threadIdx.x + off];
        __syncthreads();
    }
    if (threadIdx.x == 0)
        out[s] = lsum[s] + red[0] + bptr[0];
}

// ---------------------------------------------------------------------------
// Launch
// ---------------------------------------------------------------------------
extern "C" void kernel_launch(void* const* d_in, const int* in_sizes, int n_in,
                              void* d_out, int out_size, void* d_ws, size_t ws_size,
                              hipStream_t stream) {
    const int*   inputs = (const int*)  d_in[0];   // (B,F)
    // d_in[1] = rows, d_in[2] = cols: unused (pair index derived analytically)
    const float* w      = (const float*)d_in[3];   // (V,)
    const float* v      = (const float*)d_in[4];   // (V,E)
    const float* b      = (const float*)d_in[5];   // scalar
    const float* ew     = (const float*)d_in[6];   // (NPAIR,)
    float* out = (float*)d_out;

    float* ws     = (float*)d_ws;
    float* level2 = ws;                            // B*NPAIR floats (~12.1 MB)
    float* lsum   = level2 + (size_t)BSZ * NPAIR;  // B floats
    float* scale  = lsum + BSZ;                    // NPAIR floats
    float* shift  = scale + NPAIR;                 // NPAIR floats

    fm_gram_kernel<<<BSZ / WAVES_PER_BLOCK, 256, 0, stream>>>(
        inputs, w, v, level2, lsum);

    fm_bnstat_kernel<<<(NPAIR + 255) / 256, 256, 0, stream>>>(
        level2, ew, scale, shift);

    fm_out_kernel<<<BSZ, 256, 0, stream>>>(
        level2, scale, shift, lsum, b, out);
}